// BlastocystAuxLoss_14001593385187
// MI455X (gfx1250) — compile-verified
//
#include <hip/hip_runtime.h>
#include <hip/hip_bf16.h>

typedef float v2f __attribute__((ext_vector_type(2)));
typedef float v4f __attribute__((ext_vector_type(4)));
typedef float v8f __attribute__((ext_vector_type(8)));
typedef int   v4i __attribute__((ext_vector_type(4)));

// ws layout: ws[0] = sse accumulator (f64), ws[1] = count accumulator (f64)
__global__ void blast_init_ws(double* ws) {
    ws[0] = 0.0;
    ws[1] = 0.0;
}

__global__ __launch_bounds__(256) void blast_loss_main(
        const float* __restrict__ scores,
        const int*  __restrict__ labels,
        double* __restrict__ ws, int n) {
    const int tid      = threadIdx.x;
    const int lane     = tid & 31;          // wave32
    const int wave     = tid >> 5;          // 8 waves / block
    const int gtid     = blockIdx.x * blockDim.x + tid;
    const int nthreads = gridDim.x * blockDim.x;

    const v4f* s4 = (const v4f*)scores;
    const v4i* l4 = (const v4i*)labels;
    const int  n4 = n >> 2;

    float sse = 0.0f;
    float cnt = 0.0f;
    const float kScale = 4.0f / 7.0f;       // LATE_IDS = 8..15 -> (s-8)*4/(8-1)

    // Streaming main loop: one B128 NT load per 4 scores + one per 4 labels.
    for (int i = gtid; i < n4; i += nthreads) {
        v4f x = __builtin_nontemporal_load(&s4[i]);
        v4i s = __builtin_nontemporal_load(&l4[i]);
#pragma unroll
        for (int j = 0; j < 4; ++j) {
            int   lab = s[j];
            bool  m   = lab >= 8;
            float t   = (float)(lab - 8) * kScale;
            float d   = x[j] - t;
            sse += m ? d * d : 0.0f;
            cnt += m ? 1.0f : 0.0f;
        }
    }
    // Scalar tail (n not multiple of 4 — defensive, B=2^24 is).
    for (int i = (n4 << 2) + gtid; i < n; i += nthreads) {
        int   lab = labels[i];
        bool  m   = lab >= 8;
        float t   = (float)(lab - 8) * kScale;
        float d   = scores[i] - t;
        sse += m ? d * d : 0.0f;
        cnt += m ? 1.0f : 0.0f;
    }

    // wave32 butterfly reduction
#pragma unroll
    for (int off = 16; off > 0; off >>= 1) {
        sse += __shfl_xor(sse, off, 32);
        cnt += __shfl_xor(cnt, off, 32);
    }

    __shared__ float ls[8];
    __shared__ float lc[8];
    if (lane == 0) { ls[wave] = sse; lc[wave] = cnt; }
    __syncthreads();

    if (wave == 0) {
        // Exact f32 cross-wave reduction using V_WMMA_F32_16X16X4_F32.
        // A (16x4) layout: lanes 0-15 hold row M=lane, VGPR0=K0, VGPR1=K1;
        //                  lanes 16-31 hold row M=lane-16, VGPR0=K2, VGPR1=K3.
        // Row 0 carries sse partials, row 1 carries cnt partials; B = ones.
        // D[m][n] = sum_k A[m][k]; chain two WMMAs through C to fold all 8.
        float a0x = 0.0f, a0y = 0.0f, a1x = 0.0f, a1y = 0.0f;
        if (lane == 0)       { a0x = ls[0]; a0y = ls[1]; a1x = ls[4]; a1y = ls[5]; }
        else if (lane == 16) { a0x = ls[2]; a0y = ls[3]; a1x = ls[6]; a1y = ls[7]; }
        else if (lane == 1)  { a0x = lc[0]; a0y = lc[1]; a1x = lc[4]; a1y = lc[5]; }
        else if (lane == 17) { a0x = lc[2]; a0y = lc[3]; a1x = lc[6]; a1y = lc[7]; }
        v2f A0 = {a0x, a0y};
        v2f A1 = {a1x, a1y};
        v2f Bones = {1.0f, 1.0f};
        v8f acc = {};
        acc = __builtin_amdgcn_wmma_f32_16x16x4_f32(
                false, A0, false, Bones, (short)0, acc, false, false);
        acc = __builtin_amdgcn_wmma_f32_16x16x4_f32(
                false, A1, false, Bones, (short)0, acc, false, false);
        // D VGPR0 @ lane0 = (M=0,N=0) = block SSE; D VGPR1 @ lane0 = (M=1,N=0) = block count
        if (lane == 0) {
            atomicAdd(&ws[0], (double)acc[0]);   // global_atomic_add_f64
            atomicAdd(&ws[1], (double)acc[1]);
        }
    }
}

__global__ void blast_finalize(const double* __restrict__ ws, float* __restrict__ out) {
    double s = ws[0];
    double c = ws[1];
    double cc = c < 1.0 ? 1.0 : c;
    out[0] = (c > 0.0) ? (float)(s / cc) : 0.0f;
}

extern "C" void kernel_launch(void* const* d_in, const int* in_sizes, int n_in,
                              void* d_out, int out_size, void* d_ws, size_t ws_size,
                              hipStream_t stream) {
    const float* scores = (const float*)d_in[0];
    const int*   labels = (const int*)d_in[1];
    float*       out    = (float*)d_out;
    double*      ws     = (double*)d_ws;
    const int n = in_sizes[0];

    blast_init_ws<<<1, 1, 0, stream>>>(ws);

    // 2048 blocks x 256 threads = 524288 threads; 8 x (float4+int4) iterations
    // each -> enough MLP to saturate 23.3 TB/s HBM on a streaming reduction.
    const int blocks = 2048;
    blast_loss_main<<<blocks, 256, 0, stream>>>(scores, labels, ws, n);

    blast_finalize<<<1, 1, 0, stream>>>(ws, out);
}